// CTRGC_82892868812993
// MI455X (gfx1250) — compile-verified
//
#include <hip/hip_runtime.h>
#include <hip/hip_bf16.h>
#include <math.h>

typedef __attribute__((ext_vector_type(2))) float v2f;
typedef __attribute__((ext_vector_type(8))) float v8f;

#define NN 128
#define CC 64
#define OO 64
#define RR 8
#define TT 128
#define VV 25
#define EPSF 1e-5f
#define THALF 64        // t-chunk per WMMA block

// ---------------- K1: xm[n,c,v] = mean_t x[n,c,t,v] ----------------
__global__ void k_mean_t(const float* __restrict__ x, float* __restrict__ xm) {
    int idx = blockIdx.x * blockDim.x + threadIdx.x;
    if (idx >= NN * CC * VV) return;
    int v = idx % VV;
    int c = (idx / VV) % CC;
    int n = idx / (VV * CC);
    const float* base = x + (((size_t)n * CC + c) * TT) * VV + v;
    float s = 0.f;
    for (int t = 0; t < TT; ++t) s += base[(size_t)t * VV];
    xm[idx] = s * (1.0f / TT);
}

// ---------------- K2: x1/x2[n,r,v] = w @ xm + b ----------------
__global__ void k_x12(const float* __restrict__ xm,
                      const float* __restrict__ w1, const float* __restrict__ b1,
                      const float* __restrict__ w2, const float* __restrict__ b2,
                      float* __restrict__ x1, float* __restrict__ x2) {
    int idx = blockIdx.x * blockDim.x + threadIdx.x;
    if (idx >= NN * RR * VV) return;
    int v = idx % VV;
    int r = (idx / VV) % RR;
    int n = idx / (VV * RR);
    float s1 = b1[r], s2 = b2[r];
    const float* xb = xm + ((size_t)n * CC) * VV + v;
    for (int c = 0; c < CC; ++c) {
        float xv = xb[(size_t)c * VV];
        s1 += w1[r * CC + c] * xv;
        s2 += w2[r * CC + c] * xv;
    }
    x1[idx] = s1;
    x2[idx] = s2;
}

// ---------------- K3: rel2[n,o,u,v] = (sum_r w4*tanh(x1-x2)+b4)*alpha + A ----------------
__global__ void k_rel(const float* __restrict__ x1, const float* __restrict__ x2,
                      const float* __restrict__ w4, const float* __restrict__ b4,
                      const float* __restrict__ alpha, const float* __restrict__ A,
                      float* __restrict__ rel2) {
    int idx = blockIdx.x * blockDim.x + threadIdx.x;
    if (idx >= NN * VV * VV) return;
    int v = idx % VV;
    int u = (idx / VV) % VV;
    int n = idx / (VV * VV);
    float rl[RR];
    for (int r = 0; r < RR; ++r)
        rl[r] = tanhf(x1[((size_t)n * RR + r) * VV + u] - x2[((size_t)n * RR + r) * VV + v]);
    float al = alpha[0];
    float av = A[u * VV + v];
    for (int o = 0; o < OO; ++o) {
        float s = b4[o];
        #pragma unroll
        for (int r = 0; r < RR; ++r) s += w4[o * RR + r] * rl[r];
        rel2[((((size_t)n * OO + o) * VV) + u) * VV + v] = s * al + av;
    }
}

// ---------------- K4: x3[n,r,t,v] = w5 @ x + b5 ----------------
__global__ void k_x3(const float* __restrict__ x,
                     const float* __restrict__ w5, const float* __restrict__ b5,
                     float* __restrict__ x3) {
    int idx = blockIdx.x * blockDim.x + threadIdx.x;
    if (idx >= NN * RR * TT * VV) return;
    int v = idx % VV;
    int t = (idx / VV) % TT;
    int r = (idx / (VV * TT)) % RR;
    int n = idx / (VV * TT * RR);
    float s = b5[r];
    const float* xb = x + (((size_t)n * CC) * TT + t) * VV + v;
    const float* wb = w5 + r * CC;
    for (int c = 0; c < CC; ++c) s += wb[c] * xb[(size_t)c * TT * VV];
    x3[idx] = s;
}

// ---------------- K5: p[n,r,t] = mean_v x3 ; g[n,r] = mean_{t,v} x3 ----------------
__global__ void k_pool(const float* __restrict__ x3, float* __restrict__ p, float* __restrict__ g) {
    int nr = blockIdx.x;          // N*R blocks
    int t = threadIdx.x;          // 128 threads
    const float* base = x3 + ((size_t)nr * TT + t) * VV;
    float s = 0.f;
    for (int v = 0; v < VV; ++v) s += base[v];
    s *= (1.0f / VV);
    p[(size_t)nr * TT + t] = s;
    __shared__ float red[TT];
    red[t] = s;
    __syncthreads();
    for (int off = TT / 2; off > 0; off >>= 1) {
        if (t < off) red[t] += red[t + off];
        __syncthreads();
    }
    if (t == 0) g[nr] = red[0] * (1.0f / TT);
}

// ---------------- K6: RouteFuncMLP gate -> rf[n,r,t] ----------------
__global__ void k_gate(const float* __restrict__ p, const float* __restrict__ g,
                       const float* __restrict__ g_w, const float* __restrict__ g_b,
                       const float* __restrict__ a_w, const float* __restrict__ a_b,
                       const float* __restrict__ bn_g, const float* __restrict__ bn_b,
                       const float* __restrict__ bn_rm, const float* __restrict__ bn_rv,
                       const float* __restrict__ rf_bw, float* __restrict__ rf) {
    int n = blockIdx.x;           // N blocks
    int t = threadIdx.x;          // 128 threads
    __shared__ float gg[RR];
    __shared__ float q[RR][TT];
    __shared__ float h[RR][TT];
    if (t < RR) {
        float s = g_b[t];
        for (int i = 0; i < RR; ++i) s += g_w[t * RR + i] * g[n * RR + i];
        gg[t] = s;
    }
    __syncthreads();
    for (int r = 0; r < RR; ++r) q[r][t] = p[((size_t)n * RR + r) * TT + t] + gg[r];
    __syncthreads();
    for (int o = 0; o < RR; ++o) {
        float s = a_b[o];
        for (int i = 0; i < RR; ++i) {
            #pragma unroll
            for (int k = 0; k < 3; ++k) {
                int tt = t + k - 1;
                float qv = (tt >= 0 && tt < TT) ? q[i][tt] : 0.f;
                s += a_w[(o * RR + i) * 3 + k] * qv;
            }
        }
        s = (s - bn_rm[o]) * (bn_g[o] * rsqrtf(bn_rv[o] + EPSF)) + bn_b[o];
        h[o][t] = fmaxf(s, 0.f);
    }
    __syncthreads();
    for (int o = 0; o < RR; ++o) {
        float s = 1.0f;
        for (int i = 0; i < RR; ++i) {
            #pragma unroll
            for (int k = 0; k < 3; ++k) {
                int tt = t + k - 1;
                float hv = (tt >= 0 && tt < TT) ? h[i][tt] : 0.f;
                s += rf_bw[(o * RR + i) * 3 + k] * hv;
            }
        }
        rf[((size_t)n * RR + o) * TT + t] = s;
    }
}

// ---------------- K7: out[n,c,t,u] = sum_v rel2[n,c,u,v] * (w3@(x3*rf)+b3)[n,c,t,v] ----------------
// rf is folded into the per-lane w3 weights: X3'[t,v] = sum_r (w3[c,r]*rf[n,r,t]) * x3[n,r,t,v] + b3[c],
// so the x3 tile is staged into LDS as a PURE contiguous copy via GLOBAL_LOAD_ASYNC_TO_LDS_B128
// (ASYNCcnt path), then consumed by v_wmma_f32_16x16x4_f32.
// WMMA f32 16x16x4: A = X3'[t,v] (M=16 t, K=4 v-step), B[k=v,nn=u] = rel2[u,v].
// A layout: lane L: M = L%16, Kgpr g -> K = g + 2*(L/16). B mirrored. C/D: vgpr i -> M = i + 8*(L/16), N = L%16.
__launch_bounds__(256)
__global__ void k_out_wmma(const float* __restrict__ x3, const float* __restrict__ rf,
                           const float* __restrict__ rel2,
                           const float* __restrict__ w3, const float* __restrict__ b3,
                           float* __restrict__ out) {
    // grid.x = N * 2 * (O/8); 8 waves, one c per wave
    int blk = blockIdx.x;
    int cg = blk % (OO / 8);
    int th = (blk / (OO / 8)) % 2;
    int n = blk / ((OO / 8) * 2);
    const int t0 = th * THALF;

    __shared__ float Xs[RR * THALF * VV];   // raw x3 tile: 12800 floats = 51200 B (16B-aligned chunks)
    __shared__ float Rf[RR][THALF];         // rf tile: 512 floats

    // ---- async DMA: x3[n, r, t0..t0+63, 0..24] -> LDS (contiguous per r) ----
    unsigned lds_base = (unsigned)(size_t)(&Xs[0]);       // addrspace(3) byte offset
    const int CH_PER_R = THALF * VV / 4;                  // 400 x 16B chunks per r
    const int NCHUNK = RR * CH_PER_R;                     // 3200
    for (int i = threadIdx.x; i < NCHUNK; i += 256) {
        int r = i / CH_PER_R;
        int within = i - r * CH_PER_R;
        // global byte offset from x3 base: region start is 16B aligned (t0 in {0,64})
        unsigned goff = (unsigned)((((n * RR + r) * TT + t0) * VV) * 4 + within * 16);
        unsigned laddr = lds_base + (unsigned)i * 16u;
        asm volatile("global_load_async_to_lds_b128 %0, %1, %2"
                     :: "v"(laddr), "v"(goff), "s"(x3)
                     : "memory");
    }
    // rf tile with plain loads
    for (int i = threadIdx.x; i < RR * THALF; i += 256) {
        int r = i / THALF;
        int t = i - r * THALF;
        Rf[r][t] = rf[((size_t)n * RR + r) * TT + t0 + t];
    }
    asm volatile("s_wait_asynccnt 0x0" ::: "memory");
    __syncthreads();

    int wave = threadIdx.x >> 5;
    int lane = threadIdx.x & 31;
    int lh = lane >> 4;    // half-wave
    int ll = lane & 15;
    int c = cg * 8 + wave;

    float w3r[RR];
    #pragma unroll
    for (int r = 0; r < RR; ++r) w3r[r] = w3[c * RR + r];
    float bias = b3[c];

    // B fragments: 2 u-tiles x 7 k-steps
    v2f bfrag[2][7];
    const float* rl = rel2 + (size_t)(n * OO + c) * VV * VV;
    #pragma unroll
    for (int ut = 0; ut < 2; ++ut) {
        int u = ut * 16 + ll;
        #pragma unroll
        for (int ks = 0; ks < 7; ++ks) {
            #pragma unroll
            for (int gI = 0; gI < 2; ++gI) {
                int v = ks * 4 + gI + 2 * lh;
                bfrag[ut][ks][gI] = (u < VV && v < VV) ? rl[u * VV + v] : 0.f;
            }
        }
    }

    for (int tt = 0; tt < THALF / 16; ++tt) {
        int tA = tt * 16 + ll;                 // A-matrix row (t) owned by this lane
        float wrf[RR];
        #pragma unroll
        for (int r = 0; r < RR; ++r) wrf[r] = w3r[r] * Rf[r][tA];

        v8f acc0 = {};
        v8f acc1 = {};
        #pragma unroll
        for (int ks = 0; ks < 7; ++ks) {
            v2f a;
            #pragma unroll
            for (int gI = 0; gI < 2; ++gI) {
                int v = ks * 4 + gI + 2 * lh;
                float s = 0.f;
                if (v < VV) {
                    #pragma unroll
                    for (int r = 0; r < RR; ++r) s += wrf[r] * Xs[(r * THALF + tA) * VV + v];
                    s += bias;   // bias only on valid v (K-pad must stay exactly 0)
                }
                a[gI] = s;
            }
            acc0 = __builtin_amdgcn_wmma_f32_16x16x4_f32(false, a, false, bfrag[0][ks],
                                                         (short)0, acc0, false, false);
            acc1 = __builtin_amdgcn_wmma_f32_16x16x4_f32(false, a, false, bfrag[1][ks],
                                                         (short)0, acc1, false, false);
        }
        int tbase = t0 + tt * 16;
        #pragma unroll
        for (int i = 0; i < 8; ++i) {
            int t = tbase + i + 8 * lh;
            size_t ob = (((size_t)n * OO + c) * TT + t) * VV;
            out[ob + ll] = acc0[i];
            if (16 + ll < VV) out[ob + 16 + ll] = acc1[i];
        }
    }
}

extern "C" void kernel_launch(void* const* d_in, const int* in_sizes, int n_in,
                              void* d_out, int out_size, void* d_ws, size_t ws_size,
                              hipStream_t stream) {
    const float* x     = (const float*)d_in[0];
    const float* A     = (const float*)d_in[1];
    const float* alpha = (const float*)d_in[2];
    const float* w1    = (const float*)d_in[3];
    const float* b1    = (const float*)d_in[4];
    const float* w2    = (const float*)d_in[5];
    const float* b2    = (const float*)d_in[6];
    const float* w5    = (const float*)d_in[7];
    const float* b5    = (const float*)d_in[8];
    const float* w3    = (const float*)d_in[9];
    const float* b3    = (const float*)d_in[10];
    const float* w4    = (const float*)d_in[11];
    const float* b4    = (const float*)d_in[12];
    const float* g_w   = (const float*)d_in[13];
    const float* g_b   = (const float*)d_in[14];
    const float* a_w   = (const float*)d_in[15];
    const float* a_b   = (const float*)d_in[16];
    const float* bn_g  = (const float*)d_in[17];
    const float* bn_b  = (const float*)d_in[18];
    const float* bn_rm = (const float*)d_in[19];
    const float* bn_rv = (const float*)d_in[20];
    const float* rf_bw = (const float*)d_in[21];
    float* out = (float*)d_out;

    // workspace layout (floats)
    float* ws = (float*)d_ws;
    float* xm   = ws;                               // N*C*V       = 204800
    float* x1   = xm   + (size_t)NN * CC * VV;      // N*R*V       = 25600
    float* x2   = x1   + (size_t)NN * RR * VV;      // N*R*V       = 25600
    float* rel2 = x2   + (size_t)NN * RR * VV;      // N*O*V*V     = 5120000
    float* x3   = rel2 + (size_t)NN * OO * VV * VV; // N*R*T*V     = 3276800 (16B-aligned offset)
    float* p    = x3   + (size_t)NN * RR * TT * VV; // N*R*T       = 131072
    float* g    = p    + (size_t)NN * RR * TT;      // N*R         = 1024
    float* rf   = g    + (size_t)NN * RR;           // N*R*T       = 131072

    {
        int tot = NN * CC * VV;
        k_mean_t<<<(tot + 255) / 256, 256, 0, stream>>>(x, xm);
    }
    {
        int tot = NN * RR * VV;
        k_x12<<<(tot + 255) / 256, 256, 0, stream>>>(xm, w1, b1, w2, b2, x1, x2);
    }
    {
        int tot = NN * VV * VV;
        k_rel<<<(tot + 255) / 256, 256, 0, stream>>>(x1, x2, w4, b4, alpha, A, rel2);
    }
    {
        int tot = NN * RR * TT * VV;
        k_x3<<<(tot + 255) / 256, 256, 0, stream>>>(x, w5, b5, x3);
    }
    k_pool<<<NN * RR, TT, 0, stream>>>(x3, p, g);
    k_gate<<<NN, TT, 0, stream>>>(p, g, g_w, g_b, a_w, a_b, bn_g, bn_b, bn_rm, bn_rv, rf_bw, rf);
    k_out_wmma<<<NN * 2 * (OO / 8), 256, 0, stream>>>(x3, rf, rel2, w3, b3, out);
}